// Fusion_73057393705591
// MI455X (gfx1250) — compile-verified
//
#include <hip/hip_runtime.h>

#define DDIM 512

typedef float v2f __attribute__((ext_vector_type(2)));
typedef float v8f __attribute__((ext_vector_type(8)));

// ---------------------------------------------------------------------------
// GEMM: z[N,512] = h[N,512] @ W[512,512], fp32 WMMA 16x16x4.
// h is a virtual concat of h0 (rows [0,n0)) and h1 (rows [n0,N)); all splits
// are multiples of 16 so a 16-row tile never straddles the boundary... but we
// select per-lane anyway which is also correct.
// One wave computes a 16x64 tile: 4 accumulators share one A fragment.
// ---------------------------------------------------------------------------
__global__ __launch_bounds__(256) void gat_gemm_wmma_f32(
    const float* __restrict__ h0, int n0,
    const float* __restrict__ h1,
    const float* __restrict__ W,
    float* __restrict__ z, int n_rows)
{
    const int total_tiles = (n_rows >> 4) * (DDIM >> 6); // (N/16) * 8
    int tile = blockIdx.x * 8 + (threadIdx.x >> 5);
    if (tile >= total_tiles) return;               // wave-uniform exit (EXEC stays full)

    const int tm   = tile >> 3;
    const int tn   = tile & 7;
    const int row0 = tm << 4;
    const int col0 = tn << 6;

    const int lane = threadIdx.x & 31;
    const int half = lane >> 4;                    // 0: K pair {0,1}, 1: K pair {2,3}
    const int r    = lane & 15;                    // M for A-frag, N for B-frag

    const int arow = row0 + r;
    const float* __restrict__ Arow =
        (arow < n0) ? (h0 + (size_t)arow * DDIM)
                    : (h1 + (size_t)(arow - n0) * DDIM);

    v8f acc[4] = {};

    for (int k = 0; k < DDIM; k += 4) {
        const int ka = k + half * 2;
        v2f a;
        a.x = Arow[ka];
        a.y = Arow[ka + 1];
#pragma unroll
        for (int c = 0; c < 4; ++c) {
            v2f b;
            b.x = W[(size_t)ka       * DDIM + col0 + c * 16 + r];
            b.y = W[(size_t)(ka + 1) * DDIM + col0 + c * 16 + r];
            acc[c] = __builtin_amdgcn_wmma_f32_16x16x4_f32(
                /*neg_a=*/false, a, /*neg_b=*/false, b,
                /*c_mod=*/(short)0, acc[c],
                /*reuse_a=*/false, /*reuse_b=*/false);
        }
    }

    // C/D layout: VGPR v, lane l -> M = v + 8*(l>=16), N = l&15
#pragma unroll
    for (int c = 0; c < 4; ++c) {
#pragma unroll
        for (int v = 0; v < 8; ++v) {
            z[(size_t)(row0 + v + 8 * half) * DDIM + col0 + c * 16 + r] = acc[c][v];
        }
    }
}

// ---------------------------------------------------------------------------
// Per-node score projections: s[n] = z[n,:].a_src ; d[n] = z[n,:].a_dst
// One wave per node.
// ---------------------------------------------------------------------------
__global__ __launch_bounds__(256) void gat_proj(
    const float* __restrict__ z,
    const float* __restrict__ a_src, const float* __restrict__ a_dst,
    float* __restrict__ s, float* __restrict__ d, int n)
{
    int node = blockIdx.x * 8 + (threadIdx.x >> 5);
    if (node >= n) return;
    int lane = threadIdx.x & 31;
    const float* __restrict__ zr = z + (size_t)node * DDIM;
    float as = 0.f, ad = 0.f;
    for (int i = lane; i < DDIM; i += 32) {
        float v = zr[i];
        as += v * a_src[i];
        ad += v * a_dst[i];
    }
#pragma unroll
    for (int off = 16; off; off >>= 1) {
        as += __shfl_xor(as, off, 32);
        ad += __shfl_xor(ad, off, 32);
    }
    if (lane == 0) { s[node] = as; d[node] = ad; }
}

// ---------------------------------------------------------------------------
// Init per-graph buffers: out rows = 0, m = -inf, sum = 0
// ---------------------------------------------------------------------------
__global__ void gat_init(float* __restrict__ out, float* __restrict__ m,
                         float* __restrict__ sum, int n_nodes)
{
    size_t i = (size_t)blockIdx.x * blockDim.x + threadIdx.x;
    if (i < (size_t)n_nodes * DDIM) out[i] = 0.f;
    if (i < (size_t)n_nodes) { m[i] = -__builtin_inff(); sum[i] = 0.f; }
}

__device__ __forceinline__ void atomicMaxF(float* addr, float val)
{
    unsigned int* ua = (unsigned int*)addr;
    unsigned int old = *ua;
    while (__uint_as_float(old) < val) {
        unsigned int assumed = old;
        old = atomicCAS(ua, assumed, __float_as_uint(val));
        if (old == assumed) break;
    }
}

// Pass 1: e = leaky_relu(s[src]+d[dst], 0.01); m[dst] = max(m[dst], e)
__global__ void gat_edge1(const int* __restrict__ src, const int* __restrict__ dst,
                          const float* __restrict__ s, const float* __restrict__ d,
                          float* __restrict__ ebuf, float* __restrict__ m, int E)
{
    int e = blockIdx.x * blockDim.x + threadIdx.x;
    if (e >= E) return;
    float v = s[src[e]] + d[dst[e]];
    v = (v > 0.f) ? v : 0.01f * v;
    ebuf[e] = v;
    atomicMaxF(&m[dst[e]], v);
}

// m = where(isfinite(m), m, 0)
__global__ void gat_fixm(float* __restrict__ m, int n)
{
    int i = blockIdx.x * blockDim.x + threadIdx.x;
    if (i >= n) return;
    float v = m[i];
    if (!(fabsf(v) < __builtin_inff())) m[i] = 0.f;
}

// Pass 2: ex = exp(e - m[dst]); sum[dst] += ex
__global__ void gat_edge2(const int* __restrict__ dst,
                          float* __restrict__ ebuf, const float* __restrict__ m,
                          float* __restrict__ sum, int E)
{
    int e = blockIdx.x * blockDim.x + threadIdx.x;
    if (e >= E) return;
    float ex = expf(ebuf[e] - m[dst[e]]);
    ebuf[e] = ex;
    atomicAdd(&sum[dst[e]], ex);
}

// Pass 3: out[dst,:] += (ex/sum[dst]) * z[src,:]   (wave per edge)
__global__ __launch_bounds__(256) void gat_edge3(
    const int* __restrict__ src, const int* __restrict__ dst,
    const float* __restrict__ ebuf, const float* __restrict__ sum,
    const float* __restrict__ z, float* __restrict__ out, int E)
{
    int e = blockIdx.x * 8 + (threadIdx.x >> 5);
    if (e >= E) return;
    int lane = threadIdx.x & 31;
    int si = src[e], di = dst[e];
    float alpha = ebuf[e] / sum[di];
    const float* __restrict__ zr = z + (size_t)si * DDIM;
    float* __restrict__ orow = out + (size_t)di * DDIM;
    for (int i = lane; i < DDIM; i += 32)
        atomicAdd(&orow[i], alpha * zr[i]);
}

// ---------------------------------------------------------------------------
// Attention fusion: out = A + sum_j softmax_j(A.w_j[:512] + B_j.w_j[512:] + b_j) * B_j
// One wave per row, k <= 5 candidates.
// ---------------------------------------------------------------------------
struct FuseArgs {
    const float* B[5];
    const float* w[5];     // [1024] each
    const float* bias[5];  // [1] each
};

__global__ __launch_bounds__(256) void fuse_kernel(
    const float* __restrict__ A, FuseArgs fa, int k,
    float* __restrict__ out, int n)
{
    int row = blockIdx.x * 8 + (threadIdx.x >> 5);
    if (row >= n) return;
    int lane = threadIdx.x & 31;
    const float* __restrict__ Ar = A + (size_t)row * DDIM;

    float sc[5];
    for (int j = 0; j < k; ++j) {
        const float* __restrict__ Br = fa.B[j] + (size_t)row * DDIM;
        const float* __restrict__ w  = fa.w[j];
        float acc = 0.f;
        for (int i = lane; i < DDIM; i += 32)
            acc += Ar[i] * w[i] + Br[i] * w[DDIM + i];
#pragma unroll
        for (int off = 16; off; off >>= 1)
            acc += __shfl_xor(acc, off, 32);
        sc[j] = acc + fa.bias[j][0];
    }

    float mx = sc[0];
    for (int j = 1; j < k; ++j) mx = fmaxf(mx, sc[j]);
    float wg[5], se = 0.f;
    for (int j = 0; j < k; ++j) { wg[j] = expf(sc[j] - mx); se += wg[j]; }
    float inv = 1.f / se;
    for (int j = 0; j < k; ++j) wg[j] *= inv;

    float* __restrict__ Or = out + (size_t)row * DDIM;
    for (int i = lane; i < DDIM; i += 32) {
        float o = Ar[i];
        for (int j = 0; j < k; ++j)
            o += wg[j] * (fa.B[j] + (size_t)row * DDIM)[i];
        Or[i] = o;
    }
}

__global__ void add_kernel(const float* __restrict__ a, const float* __restrict__ b,
                           float* __restrict__ out, size_t n)
{
    size_t i = (size_t)blockIdx.x * blockDim.x + threadIdx.x;
    if (i < n) out[i] = a[i] + b[i];
}

// ---------------------------------------------------------------------------
// Host orchestration
// ---------------------------------------------------------------------------
extern "C" void kernel_launch(void* const* d_in, const int* in_sizes, int n_in,
                              void* d_out, int out_size, void* d_ws, size_t ws_size,
                              hipStream_t stream)
{
    (void)in_sizes; (void)n_in; (void)out_size; (void)ws_size;

    const float* kn   = (const float*)d_in[0];   // [384, 512]
    const float* exer = (const float*)d_in[1];   // [20000, 512]
    const float* stu  = (const float*)d_in[2];   // [50000, 512]
    const float* ukn  = (const float*)d_in[3];   // [128, 512]
    const float* bases[4] = { kn, exer, stu, ukn };

    // GRAPHS order: kk_d, kk_u, ukuk_d, ukuk_u, kuk_d, kuk_u, ukk_d, ukk_u,
    //               k_from_e, e_from_k, u_from_e, e_from_u
    static const int g_rows[12] = {384,384,128,128,512,512,512,512,20384,20384,70000,70000};
    static const int g_n0[12]   = {384,384,128,128,384,384,384,384,20000,20000,20000,20000};
    static const int g_h0[12]   = {0,0,3,3,0,0,0,0,1,1,1,1};
    static const int g_h1[12]   = {0,0,3,3,3,3,3,3,0,0,2,2};
    static const int g_E[12]    = {20000,20000,5000,5000,20000,20000,20000,20000,
                                   200000,200000,500000,500000};

    // Workspace layout (floats)
    float* ws   = (float*)d_ws;
    float* z    = ws;                                  // 70000*512
    float* gout = z + (size_t)70000 * DDIM;            // 183840*512 total
    size_t go_off[12]; size_t acc_rows = 0;
    for (int g = 0; g < 12; ++g) { go_off[g] = acc_rows * DDIM; acc_rows += (size_t)g_rows[g]; }
    float* ebuf = gout + acc_rows * DDIM;              // 500000
    float* mbuf = ebuf + 500000;                       // 70000
    float* sumb = mbuf + 70000;                        // 70000
    float* sbuf = sumb + 70000;                        // 70000
    float* dbuf = sbuf + 70000;                        // 70000

    for (int g = 0; g < 12; ++g) {
        const float* W   = (const float*)d_in[4 + 3 * g + 0];
        const float* a_s = (const float*)d_in[4 + 3 * g + 1];
        const float* a_d = (const float*)d_in[4 + 3 * g + 2];
        const int* esrc  = (const int*)d_in[62 + 2 * g + 0];
        const int* edst  = (const int*)d_in[62 + 2 * g + 1];
        const int N = g_rows[g], E = g_E[g];
        float* og = gout + go_off[g];

        int total_tiles = (N / 16) * (DDIM / 64);
        gat_gemm_wmma_f32<<<(total_tiles + 7) / 8, 256, 0, stream>>>(
            bases[g_h0[g]], g_n0[g], bases[g_h1[g]], W, z, N);

        gat_proj<<<(N + 7) / 8, 256, 0, stream>>>(z, a_s, a_d, sbuf, dbuf, N);

        unsigned init_blocks = (unsigned)(((size_t)N * DDIM + 255) / 256);
        gat_init<<<init_blocks, 256, 0, stream>>>(og, mbuf, sumb, N);

        gat_edge1<<<(E + 255) / 256, 256, 0, stream>>>(esrc, edst, sbuf, dbuf, ebuf, mbuf, E);
        gat_fixm<<<(N + 255) / 256, 256, 0, stream>>>(mbuf, N);
        gat_edge2<<<(E + 255) / 256, 256, 0, stream>>>(edst, ebuf, mbuf, sumb, E);
        gat_edge3<<<(E + 7) / 8, 256, 0, stream>>>(esrc, edst, ebuf, sumb, z, og, E);
    }

    // attn_fcs: FC_NAMES order k1..k5, uk2..uk5, e1, e2 -> d_in[40 + 2*f] (w), [.. +1] (b)
    auto fcw = [&](int f) { return (const float*)d_in[40 + 2 * f]; };
    auto fcb = [&](int f) { return (const float*)d_in[40 + 2 * f + 1]; };

    float* out_kn   = (float*)d_out;                                       // 384 rows
    float* out_exer = out_kn   + (size_t)384 * DDIM;                       // 20000 rows
    float* out_stu  = out_exer + (size_t)20000 * DDIM;                     // 50000 rows
    float* out_ukn  = out_stu  + (size_t)50000 * DDIM;                     // 128 rows

    // kn fusion: cands = [k_from_e[20000:], kk_u, kk_d, kuk_u[:384], kuk_d[:384]]
    {
        FuseArgs fa{};
        fa.B[0] = gout + go_off[8] + (size_t)20000 * DDIM;  // k_from_e tail
        fa.B[1] = gout + go_off[1];                         // kk_u
        fa.B[2] = gout + go_off[0];                         // kk_d
        fa.B[3] = gout + go_off[5];                         // kuk_u[:384]
        fa.B[4] = gout + go_off[4];                         // kuk_d[:384]
        for (int j = 0; j < 5; ++j) { fa.w[j] = fcw(j); fa.bias[j] = fcb(j); }
        fuse_kernel<<<(384 + 7) / 8, 256, 0, stream>>>(kn, fa, 5, out_kn, 384);
    }

    // ukn fusion: cands = [ukuk_u, ukuk_d, ukk_u[384:], ukk_d[384:]]
    {
        FuseArgs fa{};
        fa.B[0] = gout + go_off[3];                         // ukuk_u
        fa.B[1] = gout + go_off[2];                         // ukuk_d
        fa.B[2] = gout + go_off[7] + (size_t)384 * DDIM;    // ukk_u tail
        fa.B[3] = gout + go_off[6] + (size_t)384 * DDIM;    // ukk_d tail
        fa.B[4] = ukn;                                      // unused slot
        for (int j = 0; j < 4; ++j) { fa.w[j] = fcw(5 + j); fa.bias[j] = fcb(5 + j); }
        fa.w[4] = fcw(0); fa.bias[4] = fcb(0);              // unused slot
        fuse_kernel<<<(128 + 7) / 8, 256, 0, stream>>>(ukn, fa, 4, out_ukn, 128);
    }

    // exer fusion: cands = [e_from_u[:20000], e_from_k[:20000]]
    {
        FuseArgs fa{};
        fa.B[0] = gout + go_off[11];                        // e_from_u head
        fa.B[1] = gout + go_off[9];                         // e_from_k head
        fa.B[2] = fa.B[3] = fa.B[4] = exer;                 // unused slots
        fa.w[0] = fcw(9);  fa.bias[0] = fcb(9);             // e1
        fa.w[1] = fcw(10); fa.bias[1] = fcb(10);            // e2
        fa.w[2] = fa.w[3] = fa.w[4] = fcw(0);
        fa.bias[2] = fa.bias[3] = fa.bias[4] = fcb(0);
        fuse_kernel<<<(20000 + 7) / 8, 256, 0, stream>>>(exer, fa, 2, out_exer, 20000);
    }

    // stu_out = all_stu_emb + u_from_e[20000:]
    {
        size_t n = (size_t)50000 * DDIM;
        add_kernel<<<(unsigned)((n + 255) / 256), 256, 0, stream>>>(
            stu, gout + go_off[10] + (size_t)20000 * DDIM, out_stu, n);
    }
}